// SSVEPMamba_83631603188255
// MI455X (gfx1250) — compile-verified
//
#include <hip/hip_runtime.h>
#include <hip/hip_bf16.h>

#define L_SEQ 250
#define DMODEL 8
#define DINNER 16
#define DSTATE 16
#define DCONV 5
#define NCLS 12

typedef __attribute__((ext_vector_type(2))) float v2f;
typedef __attribute__((ext_vector_type(8))) float v8f;

__device__ __forceinline__ float sigmoidf_(float v) { return 1.0f / (1.0f + __expf(-v)); }

// One block per batch element. 256 threads = 8 wave32.
// LDS layout (floats), 16298 total = 65192 bytes:
//   s_z   [0     .. 4015]  z gate (row 250 = WMMA dump row) -> reused as y by scan
//   s_xc  [4016  .. 8015]  conv+silu output
//   s_B   [8016  .. 12031] staged x slice, then Bm (row 250 = dump row)
//   s_C   [12032 .. 16047] Cm (dump row) -> reused as out[l*8+dm] (<=2048 used)
//   s_dtr [16048 .. 16297] dt_raw[l] -> reused as FC reduction scratch (96 used)
__global__ __launch_bounds__(256) void mamba_fused_kernel(
    const float* __restrict__ x,      // (1024,1,8,250)
    const float* __restrict__ W_in,   // (32,8)
    const float* __restrict__ conv_w, // (16,1,5)
    const float* __restrict__ conv_b, // (16)
    const float* __restrict__ W_x,    // (33,16)
    const float* __restrict__ dt_w,   // (16,1)
    const float* __restrict__ dt_b,   // (16)
    const float* __restrict__ A_log,  // (16,16)
    const float* __restrict__ Dp,     // (16)
    const float* __restrict__ W_out,  // (8,16)
    const float* __restrict__ fc_w,   // (12,2000)
    const float* __restrict__ fc_b,   // (12)
    float* __restrict__ out)          // (1024,12)
{
    __shared__ float smem[16298];
    float* s_z   = smem;            // 4016 (incl. dump row 250)
    float* s_xc  = smem + 4016;     // 4000
    float* s_B   = smem + 8016;     // 4016 (x stage, then Bm)
    float* s_C   = smem + 12032;    // 4016 (Cm, then out)
    float* s_dtr = smem + 16048;    // 250  (dt_raw, then FC scratch)
    float* s_xb  = s_B;             // staged x[b] slice (2000 floats)
    float* s_y   = s_z;             // scan output aliases z (same-thread RAW only)
    float* s_out = s_C;
    float* s_red = s_dtr;

    const int b    = blockIdx.x;
    const int tid  = threadIdx.x;
    const int lane = tid & 31;
    const int wave = tid >> 5;
    const int lm   = lane & 15;   // N/M index within half-wave
    const int hi   = lane >> 4;   // K-pair select (WMMA f32 layout)
    const float* xb = x + (size_t)b * DMODEL * L_SEQ; // x[b,0,dm,l] : dm*250 + l

    // ---------------- Phase 0: stage x[b] into LDS; prefetch fc_w into caches
    for (int i = tid; i < DMODEL * L_SEQ; i += 256) s_xb[i] = xb[i];
    for (int i = tid * 32; i < NCLS * L_SEQ * DMODEL; i += 256 * 32)
        __builtin_prefetch(fc_w + i, 0, 0);
    __syncthreads();

    // ---------------- Phase 1: z = (x @ W_in^T)[:, 16:32] via v_wmma_f32_16x16x4_f32 (K=8)
    for (int mt = wave; mt < 16; mt += 8) {
        const int l0 = mt * 16;
        const int lr = min(l0 + lm, L_SEQ - 1);
        v8f acc = {};
#pragma unroll
        for (int kc = 0; kc < 8; kc += 4) {
            const int k0 = kc + 2 * hi;
            v2f a, bb;
            a.x  = s_xb[(k0    ) * L_SEQ + lr];        // A[m=l][k]
            a.y  = s_xb[(k0 + 1) * L_SEQ + lr];
            bb.x = W_in[(16 + lm) * 8 + k0];           // B[k][n] = W_in[16+n][k]
            bb.y = W_in[(16 + lm) * 8 + k0 + 1];
            acc = __builtin_amdgcn_wmma_f32_16x16x4_f32(false, a, false, bb,
                                                        (short)0, acc, false, false);
        }
#pragma unroll
        for (int r = 0; r < 8; ++r) {
            const int row = min(l0 + r + 8 * hi, L_SEQ);   // OOB lanes -> dump row 250
            s_z[row * 16 + lm] = acc[r];
        }
    }

    // ---------------- Phase 2: xc = silu(causal depthwise conv(x @ W_in^T[:, :16]) + b)
    for (int i = tid; i < L_SEQ * DINNER; i += 256) {
        const int l = i >> 4, d = i & 15;
        float accv = conv_b[d];
#pragma unroll
        for (int j = 0; j < DCONV; ++j) {
            const int t = l - (DCONV - 1) + j;
            if (t >= 0) {
                float xi = 0.f;
#pragma unroll
                for (int dm = 0; dm < DMODEL; ++dm) xi += s_xb[dm * L_SEQ + t] * W_in[d * 8 + dm];
                accv += conv_w[d * DCONV + j] * xi;
            }
        }
        s_xc[i] = accv * sigmoidf_(accv);
    }
    __syncthreads();   // xc ready; all reads of staged x done (s_B free for Bm)

    // ---------------- Phase 2b: dt_raw[l] = xc[l,:] . W_x[0,:]
    for (int l = tid; l < L_SEQ; l += 256) {
        float a = 0.f;
#pragma unroll
        for (int d = 0; d < DINNER; ++d) a += s_xc[l * 16 + d] * W_x[d];
        s_dtr[l] = a;
    }

    // ---------------- Phase 3: Bm (e=1..16) and Cm (e=17..32) via WMMA, K=16
    for (int j = wave; j < 32; j += 8) {
        const int mt = j >> 1, nt = j & 1;
        const int l0 = mt * 16;
        const int lr = min(l0 + lm, L_SEQ - 1);
        const int e0 = 1 + nt * 16;
        v8f acc = {};
#pragma unroll
        for (int kc = 0; kc < 16; kc += 4) {
            const int k0 = kc + 2 * hi;
            v2f a, bb;
            a.x  = s_xc[lr * 16 + k0];
            a.y  = s_xc[lr * 16 + k0 + 1];
            bb.x = W_x[(e0 + lm) * 16 + k0];
            bb.y = W_x[(e0 + lm) * 16 + k0 + 1];
            acc = __builtin_amdgcn_wmma_f32_16x16x4_f32(false, a, false, bb,
                                                        (short)0, acc, false, false);
        }
        float* dst = nt ? s_C : s_B;
#pragma unroll
        for (int r = 0; r < 8; ++r) {
            const int row = min(l0 + r + 8 * hi, L_SEQ);   // dump row, no EXEC games
            dst[row * 16 + lm] = acc[r];
        }
    }
    __syncthreads();

    // ---------------- Phase 4: selective scan, ALL 256 threads: lane=(d,s), 1 state/thread
    {
        const int d = tid >> 4;          // channel (wave w owns d=2w, 2w+1)
        const int s = tid & 15;          // state
        const float dtw = dt_w[d], dtb = dt_b[d], Dpd = Dp[d];
        const float As  = -__expf(A_log[d * 16 + s]);
        float h = 0.f;
        for (int l = 0; l < L_SEQ; ++l) {
            float dv = s_dtr[l] * dtw + dtb;                   // dt pre-activation
            dv = (dv > 20.f) ? dv : log1pf(__expf(dv));        // softplus
            const float xcv = s_xc[l * 16 + d];
            h = __expf(dv * As) * h + (dv * xcv) * s_B[l * 16 + s];
            float y = h * s_C[l * 16 + s];
            y += __shfl_xor(y, 1, 32);                         // reduce over 16 states
            y += __shfl_xor(y, 2, 32);
            y += __shfl_xor(y, 4, 32);
            y += __shfl_xor(y, 8, 32);
            if (s == 0) {
                y += xcv * Dpd;
                const float zv = s_z[l * 16 + d];              // same-thread read...
                y *= zv * sigmoidf_(zv);
                s_y[l * 16 + d] = y;                           // ...then overwrite (safe)
            }
        }
    }
    __syncthreads();

    // ---------------- Phase 5a: out = y @ W_out^T via WMMA (K=16, N=8 zero-padded to 16)
    for (int mt = wave; mt < 16; mt += 8) {
        const int l0 = mt * 16;
        const int lr = min(l0 + lm, L_SEQ - 1);
        v8f acc = {};
#pragma unroll
        for (int kc = 0; kc < 16; kc += 4) {
            const int k0 = kc + 2 * hi;
            v2f a, bb;
            a.x  = s_y[lr * 16 + k0];
            a.y  = s_y[lr * 16 + k0 + 1];
            bb.x = (lm < DMODEL) ? W_out[lm * 16 + k0]     : 0.f;
            bb.y = (lm < DMODEL) ? W_out[lm * 16 + k0 + 1] : 0.f;
            acc = __builtin_amdgcn_wmma_f32_16x16x4_f32(false, a, false, bb,
                                                        (short)0, acc, false, false);
        }
        if (lm < DMODEL) {   // single EXEC mask for all 8 stores; rows<=255 stay in-region
#pragma unroll
            for (int r = 0; r < 8; ++r)
                s_out[(l0 + r + 8 * hi) * 8 + lm] = acc[r];
        }
    }
    __syncthreads();

    // ---------------- Phase 5b: FC, single pass: 12 accumulators/thread, one barrier
    {
        float part[NCLS];
#pragma unroll
        for (int c = 0; c < NCLS; ++c) part[c] = 0.f;
        for (int i = tid; i < L_SEQ * DMODEL; i += 256) {
            const float v = s_out[i];
#pragma unroll
            for (int c = 0; c < NCLS; ++c) part[c] += v * fc_w[c * (L_SEQ * DMODEL) + i];
        }
#pragma unroll
        for (int c = 0; c < NCLS; ++c) {
#pragma unroll
            for (int off = 16; off > 0; off >>= 1)
                part[c] += __shfl_down(part[c], off, 32);
        }
        if (lane == 0) {
#pragma unroll
            for (int c = 0; c < NCLS; ++c) s_red[wave * NCLS + c] = part[c];
        }
        __syncthreads();
        if (tid < NCLS) {
            float sres = fc_b[tid];
#pragma unroll
            for (int w = 0; w < 8; ++w) sres += s_red[w * NCLS + tid];
            out[b * NCLS + tid] = sres;
        }
    }
}

extern "C" void kernel_launch(void* const* d_in, const int* in_sizes, int n_in,
                              void* d_out, int out_size, void* d_ws, size_t ws_size,
                              hipStream_t stream) {
    (void)in_sizes; (void)n_in; (void)out_size; (void)d_ws; (void)ws_size;
    mamba_fused_kernel<<<dim3(1024), dim3(256), 0, stream>>>(
        (const float*)d_in[0],  // x
        (const float*)d_in[1],  // W_in
        (const float*)d_in[2],  // conv_w
        (const float*)d_in[3],  // conv_b
        (const float*)d_in[4],  // W_x
        (const float*)d_in[5],  // dt_w
        (const float*)d_in[6],  // dt_b
        (const float*)d_in[7],  // A_log
        (const float*)d_in[8],  // Dp
        (const float*)d_in[9],  // W_out
        (const float*)d_in[10], // fc_w
        (const float*)d_in[11], // fc_b
        (float*)d_out);
}